// TaylorFilter_71287867179173
// MI455X (gfx1250) — compile-verified
//
#include <hip/hip_runtime.h>
#include <hip/hip_bf16.h>

typedef __attribute__((ext_vector_type(2))) float v2f;
typedef __attribute__((ext_vector_type(8))) float v8f;

#define B_      8
#define CS      64
#define CG      32
#define HW      65536
#define PIXWG   2048
#define NCHUNK  32          // HW / PIXWG
#define KC      64          // pixels per LDS sub-chunk
#define NSUB    (PIXWG/KC)
#define LDP     66          // padded LDS pitch (bank-conflict-free)

// workspace layout (float element offsets), total ~2.23 MB
#define WS_TPART 0                          // [8][32][2048] per-WG partial T
#define WS_OPART (WS_TPART + 8*32*2048)     // [8][32][64]   per-WG partial offset
#define WS_T     (WS_OPART + 8*32*64)       // [8][2048]     reduced T (g*64+s)
#define WS_OFF   (WS_T + 8*2048)            // [8][64]       reduced offset

__device__ __forceinline__ v8f wmma_f32(v2f a, v2f b, v8f c) {
  // V_WMMA_F32_16X16X4_F32: D = A(16x4) * B(4x16) + C
  return __builtin_amdgcn_wmma_f32_16x16x4_f32(false, a, false, b, (short)0, c,
                                               false, false);
}

// ---------------------------------------------------------------------------
// Pass 1: T[g,s] = sum_p guide[g,p] * (e[p]*src[s,p]),  offset[s] = sum_p e*src
// ---------------------------------------------------------------------------
__global__ __launch_bounds__(256)
void tf_pass1(const float* __restrict__ src, const float* __restrict__ guide,
              float* __restrict__ ws) {
  __shared__ float lg[CG * LDP];
  __shared__ float ls[CS * LDP];
  __shared__ float le[KC];

  const int t     = threadIdx.x;
  const int bid   = blockIdx.x;
  const int b     = bid >> 5;        // / NCHUNK
  const int chunk = bid & 31;
  const int pix0  = chunk * PIXWG;

  const int p    = t & 63;
  const int q    = t >> 6;           // 0..3
  const int wave = t >> 5;
  const int lane = t & 31;
  const int ln   = lane & 15;
  const int hi   = lane >> 4;        // 0 or 1 (half-wave)
  const int gt   = wave >> 2;        // guide tile 0..1
  const int st   = wave & 3;         // src tile 0..3

  const float* gbase = guide + (size_t)b * CG * HW + pix0;
  const float* sbase = src   + (size_t)b * CS * HW + pix0;

  v8f accT = {};
  v8f accO = {};
  v2f one; one[0] = 1.0f; one[1] = 1.0f;

  const int ar = gt * 16 + ln;       // A row  (guide channel)
  const int sc = st * 16 + ln;       // B col  (src channel)

  for (int scn = 0; scn < NSUB; ++scn) {
    __syncthreads();                 // previous iteration's LDS reads done
    const int poff = scn * KC;
    #pragma unroll
    for (int i = 0; i < 8; ++i) {    // guide: 32 ch x 64 pix, coalesced
      const int g = q + 4 * i;
      lg[g * LDP + p] = gbase[(size_t)g * HW + poff + p];
    }
    #pragma unroll
    for (int i = 0; i < 16; ++i) {   // src: 64 ch x 64 pix, coalesced
      const int s = q + 4 * i;
      ls[s * LDP + p] = sbase[(size_t)s * HW + poff + p];
    }
    __syncthreads();
    if (t < KC) {                    // e[p] = exp(-sum_g g^2)
      float ssq = 0.0f;
      #pragma unroll
      for (int g = 0; g < CG; ++g) { const float v = lg[g * LDP + t]; ssq += v * v; }
      le[t] = __expf(-ssq);
    }
    __syncthreads();
    #pragma unroll
    for (int j = 0; j < 16; ++j) {   // K = 64 pixels in steps of 4
      const int k0 = 4 * j + 2 * hi;
      v2f a;  a[0]  = lg[ar * LDP + k0];
              a[1]  = lg[ar * LDP + k0 + 1];
      v2f bb; bb[0] = ls[sc * LDP + k0]     * le[k0];
              bb[1] = ls[sc * LDP + k0 + 1] * le[k0 + 1];
      accT = wmma_f32(a,   bb, accT);  // interaction tile
      accO = wmma_f32(one, bb, accO);  // ones-row trick -> column sums
    }
  }

  // write per-WG partials (deterministic; no atomics)
  float* Tp = ws + WS_TPART + (size_t)(b * NCHUNK + chunk) * 2048;
  const int gRow = gt * 16 + 8 * hi;
  #pragma unroll
  for (int r = 0; r < 8; ++r) Tp[(gRow + r) * 64 + sc] = accT[r];
  if (gt == 0 && hi == 0)
    ws[WS_OPART + (size_t)(b * NCHUNK + chunk) * 64 + sc] = accO[0];
}

// ---------------------------------------------------------------------------
// Reduce 32 chunk partials -> T, offset
// ---------------------------------------------------------------------------
__global__ __launch_bounds__(256)
void tf_reduce(float* __restrict__ ws) {
  const int gid = blockIdx.x * 256 + threadIdx.x;
  if (gid < 8 * 2048) {
    const int b = gid >> 11, i = gid & 2047;
    float s = 0.0f;
    for (int c = 0; c < NCHUNK; ++c)
      s += ws[WS_TPART + (size_t)(b * NCHUNK + c) * 2048 + i];
    ws[WS_T + b * 2048 + i] = s;
  } else if (gid < 8 * 2048 + 8 * 64) {
    const int k = gid - 8 * 2048;
    const int b = k >> 6, i = k & 63;
    float s = 0.0f;
    for (int c = 0; c < NCHUNK; ++c)
      s += ws[WS_OPART + (size_t)(b * NCHUNK + c) * 64 + i];
    ws[WS_OFF + b * 64 + i] = s;
  }
}

// ---------------------------------------------------------------------------
// Pass 2: out[s,p] = e[p] * (offset[s] + sum_g (2*T[g,s]) * guide[g,p])
// ---------------------------------------------------------------------------
__global__ __launch_bounds__(256)
void tf_pass2(const float* __restrict__ guide, const float* __restrict__ ws,
              float* __restrict__ out) {
  __shared__ float lT[2048];
  __shared__ float lo[64];

  const int t     = threadIdx.x;
  const int bid   = blockIdx.x;
  const int b     = bid >> 5;
  const int chunk = bid & 31;
  const int pix0  = chunk * PIXWG;
  const int wave  = t >> 5;
  const int lane  = t & 31;
  const int ln    = lane & 15;
  const int hi    = lane >> 4;

  for (int i = t; i < 2048; i += 256) lT[i] = ws[WS_T + b * 2048 + i];
  if (t < 64) lo[t] = ws[WS_OFF + b * 64 + t];
  __syncthreads();

  // A fragments: A[s][k] = 2*T[g,s], fully register-resident
  v2f af[4][8];
  #pragma unroll
  for (int stt = 0; stt < 4; ++stt)
    #pragma unroll
    for (int j = 0; j < 8; ++j) {
      const int g = 4 * j + 2 * hi;
      const int s = stt * 16 + ln;
      v2f a; a[0] = 2.0f * lT[g * 64 + s];
             a[1] = 2.0f * lT[(g + 1) * 64 + s];
      af[stt][j] = a;
    }
  float offv[4][8];
  #pragma unroll
  for (int stt = 0; stt < 4; ++stt)
    #pragma unroll
    for (int r = 0; r < 8; ++r) offv[stt][r] = lo[stt * 16 + r + 8 * hi];

  const float* gb = guide + (size_t)b * CG * HW;
  const v8f zero = {};

  for (int it = 0; it < PIXWG / (16 * 8); ++it) {  // 16 pixel-tiles per wave
    const int p = pix0 + 16 * (wave + 8 * it) + ln;
    v8f acc[4];
    acc[0] = zero; acc[1] = zero; acc[2] = zero; acc[3] = zero;
    float ssq = 0.0f;
    #pragma unroll
    for (int j = 0; j < 8; ++j) {                  // K = Cg in steps of 4
      const int g = 4 * j + 2 * hi;
      const float b0 = gb[(size_t)g * HW + p];
      const float b1 = gb[(size_t)(g + 1) * HW + p];
      ssq += b0 * b0 + b1 * b1;                    // fold exp argument in
      v2f bb; bb[0] = b0; bb[1] = b1;
      #pragma unroll
      for (int stt = 0; stt < 4; ++stt)
        acc[stt] = wmma_f32(af[stt][j], bb, acc[stt]);
    }
    // lanes L and L+16 each hold half of sum_g g^2 for the same pixel
    const float e = __expf(-(ssq + __shfl_xor(ssq, 16, 32)));
    #pragma unroll
    for (int stt = 0; stt < 4; ++stt) {
      float* ob = out + ((size_t)b * CS + stt * 16 + 8 * hi) * HW + p;
      #pragma unroll
      for (int r = 0; r < 8; ++r)
        ob[(size_t)r * HW] = e * (offv[stt][r] + acc[stt][r]);
    }
  }
}

// ---------------------------------------------------------------------------
extern "C" void kernel_launch(void* const* d_in, const int* in_sizes, int n_in,
                              void* d_out, int out_size, void* d_ws, size_t ws_size,
                              hipStream_t stream) {
  (void)in_sizes; (void)n_in; (void)out_size; (void)ws_size;
  const float* src   = (const float*)d_in[0];
  const float* guide = (const float*)d_in[1];
  float* out = (float*)d_out;
  float* ws  = (float*)d_ws;

  tf_pass1<<<dim3(B_ * NCHUNK), dim3(256), 0, stream>>>(src, guide, ws);
  tf_reduce<<<dim3((8 * 2048 + 8 * 64 + 255) / 256), dim3(256), 0, stream>>>(ws);
  tf_pass2<<<dim3(B_ * NCHUNK), dim3(256), 0, stream>>>(guide, ws, out);
}